// NLinearMemoryEfficient_76888504533197
// MI455X (gfx1250) — compile-verified
//
#include <hip/hip_runtime.h>
#include <stdint.h>

typedef __attribute__((ext_vector_type(2))) float v2f;
typedef __attribute__((ext_vector_type(4))) float v4f;
typedef __attribute__((ext_vector_type(8))) float v8f;
typedef __attribute__((ext_vector_type(4))) unsigned int u32x4;
typedef __attribute__((ext_vector_type(8))) int i32x8;
typedef __attribute__((ext_vector_type(4))) int i32x4;

#define N_MAT   128
#define D_INX   64
#define D_OUTX  64
#define ROWS_PER_WG 128
#define LDS_PAD 68   // floats per LDS row: 64 data + 4 pad (bank-conflict-free frag reads)

#if __has_builtin(__builtin_amdgcn_tensor_load_to_lds) && __has_builtin(__builtin_amdgcn_s_wait_tensorcnt)
#define USE_TDM 1
#else
#define USE_TDM 0
#endif

#if USE_TDM
// Issue one TDM 2D tile load: global (row-major, row_stride_elems between rows)
// -> LDS at lds_addr, with HW padding of 4 DWORDs after every 64 DWORDs stored
// (reproduces the 68-float LDS row pitch).
__device__ __forceinline__ void tdm_load_2d(uint32_t lds_addr, const float* gptr,
                                            uint32_t tile_rows, uint32_t row_elems,
                                            uint32_t row_stride_elems, uint32_t tensor_rows) {
    const uint64_t ga = (uint64_t)(uintptr_t)gptr;
    u32x4 g0;
    g0[0] = 1u;                                                  // count=1, user descriptor
    g0[1] = lds_addr;                                            // D#.lds_addr (bytes)
    g0[2] = (uint32_t)ga;                                        // global_addr[31:0]
    g0[3] = ((uint32_t)(ga >> 32) & 0x01FFFFFFu) | (2u << 30);   // global_addr[56:32], type=2
    i32x8 g1;
    g1[0] = (int)((2u << 16)      // data_size = 4B
                | (1u << 20)      // pad_enable
                | (5u << 22)      // pad_interval: every 64 DWORDs
                | (3u << 25));    // pad_amount: 4 DWORDs
    g1[1] = (int)((row_elems & 0xFFFFu) << 16);                           // tensor_dim0[15:0]
    g1[2] = (int)(((row_elems >> 16) & 0xFFFFu)
                | ((tensor_rows & 0xFFFFu) << 16));                       // dim0 hi | tensor_dim1 lo
    g1[3] = (int)(((tensor_rows >> 16) & 0xFFFFu)
                | ((row_elems & 0xFFFFu) << 16));                         // dim1 hi | tile_dim0
    g1[4] = (int)(tile_rows & 0xFFFFu);                                   // tile_dim1, tile_dim2=0
    g1[5] = (int)row_stride_elems;                                        // tensor_dim0_stride[31:0]
    g1[6] = 0;                                                            // stride hi, dim1_stride lo
    g1[7] = 0;
    const i32x4 z4 = {0, 0, 0, 0};
#if defined(__clang_major__) && (__clang_major__ >= 23)
    const i32x8 z8 = {0, 0, 0, 0, 0, 0, 0, 0};
    __builtin_amdgcn_tensor_load_to_lds(g0, g1, z4, z4, z8, 0);
#else
    __builtin_amdgcn_tensor_load_to_lds(g0, g1, z4, z4, 0);
#endif
}
#endif

__global__ __launch_bounds__(256)
void nlinear_wmma_f32(const float* __restrict__ x,
                      const float* __restrict__ W,
                      const float* __restrict__ bias,
                      float* __restrict__ out) {
    __shared__ __align__(16) float sW[D_OUTX * LDS_PAD];       // W[n]: [o][k], padded
    __shared__ __align__(16) float sX[ROWS_PER_WG * LDS_PAD];  // x tile: [row][k], padded

    const int n        = blockIdx.y;
    const int row_base = blockIdx.x * ROWS_PER_WG;
    const int tid      = threadIdx.x;
    const int lane     = tid & 31;
    const int wave     = tid >> 5;       // 0..7
    const int col      = lane & 15;      // N-index / M-index within fragment
    const int hi       = lane >> 4;      // half-wave select (K split)

#if USE_TDM
    // ---- wave 0 DMAs both tiles into LDS via the Tensor Data Mover ----
    if (tid < 32) {
        // flat LDS pointer low 32 bits == wave-relative LDS byte offset (ISA 10.2)
        tdm_load_2d((uint32_t)(uintptr_t)&sW[0],
                    W + (size_t)n * (D_OUTX * D_INX),
                    /*tile_rows=*/D_OUTX, /*row_elems=*/D_INX,
                    /*row_stride=*/D_INX, /*tensor_rows=*/D_OUTX);
        tdm_load_2d((uint32_t)(uintptr_t)&sX[0],
                    x + ((size_t)row_base * N_MAT + n) * D_INX,
                    /*tile_rows=*/ROWS_PER_WG, /*row_elems=*/D_INX,
                    /*row_stride=*/N_MAT * D_INX, /*tensor_rows=*/ROWS_PER_WG);
        __builtin_amdgcn_s_wait_tensorcnt(0);
    }
#else
    // ---- fallback: cooperative load W[n] (64x64 f32, 16KB) into LDS via b128 ----
    {
        const float* wg = W + (size_t)n * (D_OUTX * D_INX);
        const int o   = tid >> 2;   // 0..63 output row
        const int seg = tid & 3;    // 16-float segment
        const v4f* src = (const v4f*)(wg + o * D_INX + seg * 16);
        v4f*       dst = (v4f*)(sW + o * LDS_PAD + seg * 16);
#pragma unroll
        for (int j = 0; j < 4; ++j) dst[j] = src[j];
    }
    // ---- fallback: cooperative load x tile (128x64 f32, 32KB) into LDS via b128 ----
    {
        const float* xg = x + ((size_t)row_base * N_MAT + n) * D_INX;
        const int r    = tid >> 1;  // 0..127 tile row
        const int half = tid & 1;   // 32-float half-row
        const v4f* src = (const v4f*)(xg + (size_t)r * (N_MAT * D_INX) + half * 32);
        v4f*       dst = (v4f*)(sX + r * LDS_PAD + half * 32);
#pragma unroll
        for (int j = 0; j < 8; ++j) dst[j] = src[j];
    }
#endif
    __syncthreads();

    // ---- accumulators: 4 tiles of 16x16 f32, bias-initialized ----
    // C/D layout: VGPR r, lanes 0-15 -> M=r, N=lane; lanes 16-31 -> M=r+8, N=lane-16.
    v8f acc[4];
#pragma unroll
    for (int t = 0; t < 4; ++t) {
        const float bv = bias[n * D_OUTX + t * 16 + col];
#pragma unroll
        for (int j = 0; j < 8; ++j) acc[t][j] = bv;
    }

    // ---- K loop: 16 steps of V_WMMA_F32_16X16X4_F32 per output tile ----
    // A (16x4 MxK): lanes 0-15 row M=lane, K={kb,kb+1}; lanes 16-31 row M=lane-16, K={kb+2,kb+3}
    // B (4x16 KxN): lanes 0-15 col N=lane, K={kb,kb+1}; lanes 16-31 col N=lane-16, K={kb+2,kb+3}
    const float* aRow = sX + (wave * 16 + col) * LDS_PAD;
#pragma unroll
    for (int k = 0; k < 16; ++k) {
        const int ko = 4 * k + 2 * hi;
        const v2f a = *(const v2f*)(aRow + ko);
#pragma unroll
        for (int t = 0; t < 4; ++t) {
            const v2f bfr = *(const v2f*)(sW + (t * 16 + col) * LDS_PAD + ko);
            acc[t] = __builtin_amdgcn_wmma_f32_16x16x4_f32(
                /*neg_a=*/false, a, /*neg_b=*/false, bfr,
                /*c_mod=*/(short)0, acc[t],
                /*reuse_a=*/false, /*reuse_b=*/false);
        }
    }

    // ---- store: out[b, n, o], half-wave writes 64B contiguous per (t,r) ----
    float* og = out + ((size_t)(row_base + wave * 16) * N_MAT + n) * D_OUTX;
#pragma unroll
    for (int t = 0; t < 4; ++t) {
#pragma unroll
        for (int r = 0; r < 8; ++r) {
            const int m = r + 8 * hi;
            og[(size_t)m * (N_MAT * D_OUTX) + t * 16 + col] = acc[t][r];
        }
    }
}

extern "C" void kernel_launch(void* const* d_in, const int* in_sizes, int n_in,
                              void* d_out, int out_size, void* d_ws, size_t ws_size,
                              hipStream_t stream) {
    const float* x    = (const float*)d_in[0];
    const float* W    = (const float*)d_in[1];
    const float* bias = (const float*)d_in[2];
    float* out        = (float*)d_out;

    const int B = 8192;
    dim3 grid(B / ROWS_PER_WG, N_MAT);  // 64 x 128 workgroups
    nlinear_wmma_f32<<<grid, 256, 0, stream>>>(x, W, bias, out);
}